// PointsRenderer_NoDistWeight_12068858102120
// MI455X (gfx1250) — compile-verified
//
#include <hip/hip_runtime.h>

// PointsRenderer_NoDistWeight forward: alpha compositing of K point fragments.
// HBM-bound (128 MB mandatory traffic, ~0.27 GFLOP) -> optimize data movement:
//  - async global->LDS double-buffered staging of idx/dists (CDNA5 ASYNCcnt path)
//  - contrib==0 gather skip (exact under the reference formula)
//  - b128 vector loads/stores, non-temporal output stores (preserve L2 for features)

#define KFRAG 8
#define CFEAT 16
#define TPB 256
#define SUBTILES 4
#define PIX_PER_BLOCK (TPB * SUBTILES)

typedef float v4f __attribute__((ext_vector_type(4)));
typedef int   v4i __attribute__((ext_vector_type(4)));

// Per-lane async copy of one pixel's 8 ints + 8 floats (32B each) into LDS.
// GVS addressing: saddr = 64-bit base SGPR pair, vaddr = 32-bit byte offset.
// INST_OFFSET applies to both the global and LDS side -> linear copy.
__device__ __forceinline__ void async_copy_pixel(const int* __restrict__ idxp,
                                                 const float* __restrict__ dstp,
                                                 int pix,
                                                 unsigned lds_idx,
                                                 unsigned lds_dst) {
  int voff = pix * (KFRAG * 4);  // byte offset of this pixel's fragment block
  asm volatile(
      "global_load_async_to_lds_b128 %0, %2, %3\n\t"
      "global_load_async_to_lds_b128 %0, %2, %3 offset:16\n\t"
      "global_load_async_to_lds_b128 %1, %2, %4\n\t"
      "global_load_async_to_lds_b128 %1, %2, %4 offset:16"
      :
      : "v"(lds_idx), "v"(lds_dst), "v"(voff), "s"(idxp), "s"(dstp)
      : "memory");
}

__global__ void __launch_bounds__(TPB)
points_composite_kernel(const int* __restrict__ idx,
                        const float* __restrict__ dists,
                        const float* __restrict__ feats,
                        float* __restrict__ out,
                        int npix) {
  __shared__ int   s_idx[2][TPB * KFRAG];
  __shared__ float s_dst[2][TPB * KFRAG];

  const int t = threadIdx.x;
  const int pix_base = blockIdx.x * PIX_PER_BLOCK + t;

  // Raw LDS byte addresses (low 32 bits of the generic pointer == ds offset).
  unsigned li[2], ld[2];
  li[0] = (unsigned)(uintptr_t)&s_idx[0][t * KFRAG];
  li[1] = (unsigned)(uintptr_t)&s_idx[1][t * KFRAG];
  ld[0] = (unsigned)(uintptr_t)&s_dst[0][t * KFRAG];
  ld[1] = (unsigned)(uintptr_t)&s_dst[1][t * KFRAG];

  // Prologue: kick off sub-tile 0 (clamped address for tail lanes; results unused).
  {
    int p = pix_base;
    if (p > npix - 1) p = npix - 1;
    if (p < 0) p = 0;
    async_copy_pixel(idx, dists, p, li[0], ld[0]);
  }

#pragma unroll
  for (int s = 0; s < SUBTILES; ++s) {
    const int buf = s & 1;
    if (s + 1 < SUBTILES) {
      // LDS slot for tile s+1 was last read in iteration s-1; its ds reads are
      // long done, but be explicit before the async engine overwrites LDS.
      asm volatile("s_wait_dscnt 0x0" ::: "memory");
      int p = pix_base + (s + 1) * TPB;
      if (p > npix - 1) p = npix - 1;
      if (p < 0) p = 0;
      async_copy_pixel(idx, dists, p, li[(s + 1) & 1], ld[(s + 1) & 1]);
      // 4 loads outstanding for tile s+1; wait until tile s's 4 are complete.
      asm volatile("s_wait_asynccnt 0x4" ::: "memory");
    } else {
      asm volatile("s_wait_asynccnt 0x0" ::: "memory");
    }
    asm volatile("" ::: "memory");  // compiler barrier: LDS now holds tile s

    const int pix = pix_base + s * TPB;
    if (pix < npix) {
      const int*   si = &s_idx[buf][t * KFRAG];
      const float* sd = &s_dst[buf][t * KFRAG];
      v4i i0 = *(const v4i*)(si);
      v4i i1 = *(const v4i*)(si + 4);
      v4f d0 = *(const v4f*)(sd);
      v4f d1 = *(const v4f*)(sd + 4);

      const float r2 = 0.0125f * 0.0125f;
      float T = 1.0f;
      v4f a0 = 0.0f, a1 = 0.0f, a2 = 0.0f, a3 = 0.0f;

#pragma unroll
      for (int k = 0; k < KFRAG; ++k) {
        const int   id = (k < 4) ? i0[k] : i1[k - 4];
        const float d  = (k < 4) ? d0[k] : d1[k - 4];
        float w = (d > 0.0f) ? 1.0f : (1.0f - d / r2);
        if (id < 0) w = 0.0f;
        const float contrib = w * T;
        T = T * (1.0f - w);
        // Exact skip: zero-contribution fragments add nothing. Under the
        // reference formula this leaves ~1 gather per pixel.
        if (contrib != 0.0f) {
          const v4f* f = (const v4f*)(feats + (long long)id * CFEAT);
          v4f f0 = f[0], f1 = f[1], f2 = f[2], f3 = f[3];
          a0 += contrib * f0;
          a1 += contrib * f1;
          a2 += contrib * f2;
          a3 += contrib * f3;
        }
      }

      v4f* o = (v4f*)(out + (long long)pix * CFEAT);
      __builtin_nontemporal_store(a0, o + 0);
      __builtin_nontemporal_store(a1, o + 1);
      __builtin_nontemporal_store(a2, o + 2);
      __builtin_nontemporal_store(a3, o + 3);
    }
  }
}

extern "C" void kernel_launch(void* const* d_in, const int* in_sizes, int n_in,
                              void* d_out, int out_size, void* d_ws, size_t ws_size,
                              hipStream_t stream) {
  (void)n_in; (void)d_ws; (void)ws_size; (void)out_size;
  const int*   idx   = (const int*)d_in[0];
  const float* dists = (const float*)d_in[1];
  const float* feats = (const float*)d_in[2];
  float*       out   = (float*)d_out;

  const int npix   = in_sizes[0] / KFRAG;           // N*H*W
  const int blocks = (npix + PIX_PER_BLOCK - 1) / PIX_PER_BLOCK;

  hipLaunchKernelGGL(points_composite_kernel, dim3(blocks), dim3(TPB), 0, stream,
                     idx, dists, feats, out, npix);
}